// PhiRestraint_81612968558778
// MI455X (gfx1250) — compile-verified
//
#include <hip/hip_runtime.h>
#include <stdint.h>

#define BB    4
#define LL    768
#define PP    200000
#define NB    15
#define NBLK  384          // multiple of 32 (kernel2 relies on it)
#define NTHR  256          // 8 waves of 32

typedef float v2f __attribute__((ext_vector_type(2)));
typedef float v8f __attribute__((ext_vector_type(8)));

// -------- stage-1: per-block partial sums ---------------------------------
__global__ __launch_bounds__(NTHR)
void phi_partial_kernel(const float* __restrict__ CA,
                        const float* __restrict__ CB,
                        const float* __restrict__ coeff,
                        const float* __restrict__ cutoffs,
                        const int*   __restrict__ x_idx,
                        const int*   __restrict__ y_idx,
                        float*       __restrict__ partials)
{
    __shared__ __align__(16) float sCA[BB * LL * 3];   // 36,864 B
    __shared__ __align__(16) float sCB[BB * LL * 3];   // 36,864 B
    __shared__ float swsum[NTHR / 32];

    const int tid = threadIdx.x;

    // ---- async DMA of CA/CB into LDS (ASYNCcnt path, b128 chunks) --------
    {
        const uint32_t caBase = (uint32_t)(uintptr_t)(&sCA[0]); // low 32b of flat LDS ptr = LDS offset
        const uint32_t cbBase = (uint32_t)(uintptr_t)(&sCB[0]);
#pragma unroll
        for (int j = 0; j < (BB * LL * 3 * 4 / 16) / NTHR; ++j) {   // 9 chunks/thread/array
            const uint32_t goff = (uint32_t)(tid + j * NTHR) * 16u;
            const uint32_t la = caBase + goff;
            const uint32_t lb = cbBase + goff;
            asm volatile("global_load_async_to_lds_b128 %0, %1, %2"
                         :: "v"(la), "v"(goff), "s"(CA) : "memory");
            asm volatile("global_load_async_to_lds_b128 %0, %1, %2"
                         :: "v"(lb), "v"(goff), "s"(CB) : "memory");
        }
    }
    asm volatile("s_wait_asynccnt 0x0" ::: "memory");
    __syncthreads();

    // cutoffs is a linspace: bin = floor((phi - c0)/d), t = phi - (c0 + bin*d)
    const float c0    = cutoffs[0];
    const float dcut  = cutoffs[1] - c0;
    const float inv_d = 1.0f / dcut;

    float acc = 0.0f;

    for (int p = blockIdx.x * NTHR + tid; p < PP; p += NBLK * NTHR) {
        const int x = x_idx[p];
        const int y = y_idx[p];
        const float* crow = coeff + ((size_t)x * LL + y) * (NB * 4);  // 240 B row, 16B aligned
        __builtin_prefetch(crow, 0, 1);   // global_prefetch_b8: fetch row while acos chain runs
#pragma unroll
        for (int b = 0; b < BB; ++b) {    // batch-inner: the 4 hits share this row's cachelines
            const int xo = (b * LL + x) * 3;
            const int yo = (b * LL + y) * 3;
            const float ax = sCA[xo + 0], ay = sCA[xo + 1], az = sCA[xo + 2];
            const float bx = sCB[xo + 0], by = sCB[xo + 1], bz = sCB[xo + 2];
            const float cx = sCB[yo + 0], cy = sCB[yo + 1], cz = sCB[yo + 2];
            const float u1x = ax - bx, u1y = ay - by, u1z = az - bz;
            const float u2x = cx - bx, u2y = cy - by, u2z = cz - bz;
            const float dot = u1x * u2x + u1y * u2y + u1z * u2z;
            const float n1  = sqrtf(u1x * u1x + u1y * u1y + u1z * u1z);
            const float n2  = sqrtf(u2x * u2x + u2y * u2y + u2z * u2z);
            float cphi = dot / (n1 * n2 + 1e-8f);
            cphi = fminf(fmaxf(cphi, -1.0f + 1e-7f), 1.0f - 1e-7f);
            const float phi = acosf(cphi);
            int bin = (int)floorf((phi - c0) * inv_d);
            bin = bin < 0 ? 0 : (bin > NB - 1 ? NB - 1 : bin);
            const float t = phi - (c0 + (float)bin * dcut);
            const float4 cs = *reinterpret_cast<const float4*>(crow + bin * 4);
            acc += ((cs.w * t + cs.z) * t + cs.y) * t + cs.x;
        }
    }

    // wave32 tree reduce -> per-block partial (deterministic, no atomics)
#pragma unroll
    for (int off = 16; off > 0; off >>= 1) acc += __shfl_down(acc, off, 32);
    if ((tid & 31) == 0) swsum[tid >> 5] = acc;
    __syncthreads();
    if (tid == 0) {
        float s = 0.0f;
#pragma unroll
        for (int w = 0; w < NTHR / 32; ++w) s += swsum[w];
        partials[blockIdx.x] = s;
    }
}

// -------- stage-2: single-wave deterministic reduce (WMMA cross-lane sum) --
__global__ __launch_bounds__(32)
void phi_reduce_kernel(const float* __restrict__ partials, float* __restrict__ out)
{
    const int lane = threadIdx.x;           // 32 lanes, EXEC all-ones (WMMA requirement)
    float s = 0.0f;
#pragma unroll
    for (int i = 0; i < NBLK / 32; ++i) s += partials[lane + i * 32];

    // D = ones(16x4) * B(4x16) + 0 : every row of D is the per-column sum of B.
    // B vgpr0 = this lane's partial, vgpr1 = 0 -> sum of D row 0 == sum of all 32 lanes,
    // independent of the exact lane<->(K,N) mapping since A is all-ones.
    v2f a;  a[0] = 1.0f; a[1] = 1.0f;
    v2f bm; bm[0] = s;   bm[1] = 0.0f;
    v8f c = {};
    v8f d = __builtin_amdgcn_wmma_f32_16x16x4_f32(false, a, false, bm,
                                                  (short)0, c, false, false);
    float col = d[0];                        // lane l holds colsum_{l&15} (rows identical)
#pragma unroll
    for (int off = 8; off > 0; off >>= 1) col += __shfl_down(col, off, 32);
    if (lane == 0) out[0] = col;
}

// -------- launcher ---------------------------------------------------------
extern "C" void kernel_launch(void* const* d_in, const int* in_sizes, int n_in,
                              void* d_out, int out_size, void* d_ws, size_t ws_size,
                              hipStream_t stream)
{
    (void)in_sizes; (void)n_in; (void)out_size; (void)ws_size;
    const float* CA      = (const float*)d_in[0];
    const float* CB      = (const float*)d_in[1];
    const float* coeff   = (const float*)d_in[2];
    const float* cutoffs = (const float*)d_in[3];
    const int*   x_idx   = (const int*)d_in[4];
    const int*   y_idx   = (const int*)d_in[5];
    float* partials = (float*)d_ws;          // NBLK floats of scratch

    phi_partial_kernel<<<NBLK, NTHR, 0, stream>>>(CA, CB, coeff, cutoffs,
                                                  x_idx, y_idx, partials);
    phi_reduce_kernel<<<1, 32, 0, stream>>>(partials, (float*)d_out);
}